// RAA_23579370455160
// MI455X (gfx1250) — compile-verified
//
#include <hip/hip_runtime.h>
#include <math.h>

typedef float v2f __attribute__((ext_vector_type(2)));
typedef float v8f __attribute__((ext_vector_type(8)));

#define N1 8192
#define N2 4096
#define KD 8

// workspace layout (in floats)
#define OFF_M   0
#define OFF_ZI  64
#define OFF_ZJ  (OFF_ZI + N1*KD)          // 64 + 65536
#define OFF_SI  (OFF_ZJ + N2*KD)          // + 32768  (float2 per i row)
#define OFF_SJ  (OFF_SI + 2*N1)           // + 16384  (float2 per j row)

// ---------------------------------------------------------------------------
// Kernel 1: M = softmax(Z, axis=1) @ softmax(C, axis=1)   (8x8), zero d_out
// ---------------------------------------------------------------------------
__global__ __launch_bounds__(256) void k_mix(const float* __restrict__ Z,
                                             const float* __restrict__ C,
                                             float* __restrict__ Mout,
                                             float* __restrict__ out)
{
    __shared__ float red[256];
    __shared__ float maxZ[8];
    __shared__ float sumZ[8];
    __shared__ float Msh[64];
    const int tid = threadIdx.x;

    // pass 1: per-row max of Z (8 rows x 8192)
    float pm[8];
#pragma unroll
    for (int p = 0; p < 8; ++p) pm[p] = -3.4e38f;
    for (int n = tid; n < N1; n += 256) {
#pragma unroll
        for (int p = 0; p < 8; ++p) pm[p] = fmaxf(pm[p], Z[p * N1 + n]);
    }
    for (int p = 0; p < 8; ++p) {
        red[tid] = pm[p];
        __syncthreads();
        for (int s = 128; s > 0; s >>= 1) {
            if (tid < s) red[tid] = fmaxf(red[tid], red[tid + s]);
            __syncthreads();
        }
        if (tid == 0) maxZ[p] = red[0];
        __syncthreads();
    }

    // pass 2: per-row sum of exp(Z - max)
    float ps[8];
#pragma unroll
    for (int p = 0; p < 8; ++p) ps[p] = 0.f;
    for (int n = tid; n < N1; n += 256) {
#pragma unroll
        for (int p = 0; p < 8; ++p) ps[p] += __expf(Z[p * N1 + n] - maxZ[p]);
    }
    for (int p = 0; p < 8; ++p) {
        red[tid] = ps[p];
        __syncthreads();
        for (int s = 128; s > 0; s >>= 1) {
            if (tid < s) red[tid] += red[tid + s];
            __syncthreads();
        }
        if (tid == 0) sumZ[p] = red[0];
        __syncthreads();
    }

    // pass 3: accumulate M[p][q] = sum_n softmaxZ[p][n] * softmaxC[n][q]
    float Macc[64];
#pragma unroll
    for (int e = 0; e < 64; ++e) Macc[e] = 0.f;
    for (int n = tid; n < N1; n += 256) {
        float c[8];
        float cmax = -3.4e38f;
#pragma unroll
        for (int q = 0; q < 8; ++q) { c[q] = C[n * 8 + q]; cmax = fmaxf(cmax, c[q]); }
        float csum = 0.f;
#pragma unroll
        for (int q = 0; q < 8; ++q) { c[q] = __expf(c[q] - cmax); csum += c[q]; }
        const float cinv = 1.f / csum;
        float wz[8];
#pragma unroll
        for (int p = 0; p < 8; ++p) wz[p] = __expf(Z[p * N1 + n] - maxZ[p]) / sumZ[p];
#pragma unroll
        for (int p = 0; p < 8; ++p)
#pragma unroll
            for (int q = 0; q < 8; ++q)
                Macc[p * 8 + q] += wz[p] * (c[q] * cinv);
    }
    if (tid < 64) Msh[tid] = 0.f;
    __syncthreads();
#pragma unroll
    for (int e = 0; e < 64; ++e) atomicAdd(&Msh[e], Macc[e]);
    __syncthreads();
    if (tid < 64) Mout[tid] = Msh[tid];
    if (tid == 0) out[0] = 0.f;
}

// ---------------------------------------------------------------------------
// Kernel 2: z = lat @ M (rows x 8); stat = (||z||^2 + sgn*2e-6*sum(z), beta/gamma)
// ---------------------------------------------------------------------------
__global__ __launch_bounds__(256) void k_latent(const float* __restrict__ lat,
                                                const float* __restrict__ bg,
                                                const float* __restrict__ Msrc,
                                                float* __restrict__ zout,
                                                float2* __restrict__ stat,
                                                float sgn, int nrows)
{
    __shared__ float Msh[64];
    if (threadIdx.x < 64) Msh[threadIdx.x] = Msrc[threadIdx.x];
    __syncthreads();
    const int r = blockIdx.x * 256 + threadIdx.x;
    if (r >= nrows) return;

    float l[8];
#pragma unroll
    for (int p = 0; p < 8; ++p) l[p] = lat[r * 8 + p];
    float s1 = 0.f, s2 = 0.f;
#pragma unroll
    for (int q = 0; q < 8; ++q) {
        float s = 0.f;
#pragma unroll
        for (int p = 0; p < 8; ++p) s += l[p] * Msh[p * 8 + q];
        zout[r * 8 + q] = s;
        s1 += s;
        s2 += s * s;
    }
    stat[r] = make_float2(s2 + sgn * 2e-6f * s1, bg[r]);
}

// ---------------------------------------------------------------------------
// Kernel 3: main pairwise reduction.
// One wave per 16-row i-tile; zi A-fragment reused across 16 j-tiles.
// dot(zi_i, zj_j) via 2x v_wmma_f32_16x16x4_f32; transcendental epilogue
// uses raw v_sqrt_f32 / v_exp_f32 / v_log_f32 (no IEEE refinement chains).
// ---------------------------------------------------------------------------
__global__ __launch_bounds__(256) void k_main(const int* __restrict__ A,
                                              const float2* __restrict__ stat_i,
                                              const float2* __restrict__ stat_j,
                                              const float* __restrict__ zi,
                                              const float* __restrict__ zj,
                                              const float* __restrict__ a_scalar,
                                              float* __restrict__ out)
{
    const int lane = threadIdx.x & 31;
    const int wave = threadIdx.x >> 5;
    const int half = lane >> 4;        // 0: lanes 0-15, 1: lanes 16-31
    const int lm   = lane & 15;
    const int it   = blockIdx.x * 8 + wave;     // 512 i-tiles
    const int i0   = it * 16;
    const float aa = a_scalar[0];

    // A-matrix fragment (16x4 f32, 2 VGPRs): lane holds row = lm,
    // VGPR v of chunk kk holds K = kk*4 + 2*half + v.  Loaded once, reused.
    const float* zrow = zi + (size_t)(i0 + lm) * KD;
    v2f afrag0, afrag1;
    afrag0.x = zrow[0 + 2 * half + 0];
    afrag0.y = zrow[0 + 2 * half + 1];
    afrag1.x = zrow[4 + 2 * half + 0];
    afrag1.y = zrow[4 + 2 * half + 1];

    // per-lane row stats for the 8 C-matrix rows this lane owns (M = r + 8*half)
    float ri[8], bi[8];
#pragma unroll
    for (int r = 0; r < 8; ++r) {
        float2 s = stat_i[i0 + r + 8 * half];
        ri[r] = s.x; bi[r] = s.y;
    }

    float acc = 0.f;
    for (int g = 0; g < 16; ++g) {
        const int jt = blockIdx.y * 16 + g;     // 256 j-tiles total
        const int j0 = jt * 16;

        // B-matrix fragment (4x16 f32, 2 VGPRs): lane holds col = lm,
        // VGPR v of chunk kk holds K = kk*4 + 2*half + v.
        const float* zcol = zj + (size_t)(j0 + lm) * KD;
        v2f b0, b1;
        b0.x = zcol[0 + 2 * half + 0];
        b0.y = zcol[0 + 2 * half + 1];
        b1.x = zcol[4 + 2 * half + 0];
        b1.y = zcol[4 + 2 * half + 1];

        v8f c = {};
        c = __builtin_amdgcn_wmma_f32_16x16x4_f32(false, afrag0, false, b0,
                                                  (short)0, c, false, false);
        c = __builtin_amdgcn_wmma_f32_16x16x4_f32(false, afrag1, false, b1,
                                                  (short)0, c, false, false);

        const int j = j0 + lm;
        const float2 sj = stat_j[j];
        const float rj = sj.x, gmj = sj.y;

#pragma unroll
        for (int r = 0; r < 8; ++r) {
            const int i = i0 + r + 8 * half;
            float d2 = ri[r] + rj + 8e-12f - 2.f * c[r];
            d2 = fmaxf(d2, 0.f);
            // raw hardware sqrt (single v_sqrt_f32, ~1ulp) — no IEEE fixup chain
            const float theta = bi[r] + gmj - aa * __builtin_amdgcn_sqrtf(d2);
            const int Aij = A[(size_t)i * N2 + j];
            // stable softplus = log1p(exp(theta)); |.| and - fold into v_exp src mods
            const float sp = fmaxf(theta, 0.f) +
                             __logf(1.f + __expf(-fabsf(theta)));
            acc += theta * (float)Aij - sp;
        }
    }

    __shared__ float red[256];
    red[threadIdx.x] = acc;
    __syncthreads();
    for (int s = 128; s > 0; s >>= 1) {
        if (threadIdx.x < s) red[threadIdx.x] += red[threadIdx.x + s];
        __syncthreads();
    }
    if (threadIdx.x == 0) atomicAdd(out, red[0]);
}

// ---------------------------------------------------------------------------
extern "C" void kernel_launch(void* const* d_in, const int* in_sizes, int n_in,
                              void* d_out, int out_size, void* d_ws, size_t ws_size,
                              hipStream_t stream)
{
    const int*   A     = (const int*)d_in[0];
    const float* beta  = (const float*)d_in[1];
    const float* gamma = (const float*)d_in[2];
    const float* a     = (const float*)d_in[3];
    const float* Z     = (const float*)d_in[4];
    const float* C     = (const float*)d_in[5];
    const float* lzi   = (const float*)d_in[6];
    const float* lzj   = (const float*)d_in[7];
    float* out = (float*)d_out;
    float* ws  = (float*)d_ws;

    float*  Mw = ws + OFF_M;
    float*  zi = ws + OFF_ZI;
    float*  zj = ws + OFF_ZJ;
    float2* si = (float2*)(ws + OFF_SI);
    float2* sj = (float2*)(ws + OFF_SJ);

    k_mix<<<1, 256, 0, stream>>>(Z, C, Mw, out);
    k_latent<<<N1 / 256, 256, 0, stream>>>(lzi, beta,  Mw, zi, si, +1.0f, N1);
    k_latent<<<N2 / 256, 256, 0, stream>>>(lzj, gamma, Mw, zj, sj, -1.0f, N2);

    dim3 grid(64, 16);   // 512 i-tiles (8 waves/block) x 16 j-groups (16 tiles each)
    k_main<<<grid, 256, 0, stream>>>(A, si, sj, zi, zj, a, out);
}